// OptimizationFunc_73280732004780
// MI455X (gfx1250) — compile-verified
//
#include <hip/hip_runtime.h>

// Problem constants (match reference)
#define DY    3
#define NCP   512
#define NOBS  256
#define BATCH 32

typedef float v2f __attribute__((ext_vector_type(2)));
typedef float v8f __attribute__((ext_vector_type(8)));

__device__ __forceinline__ float bperm_f(int srcLane, float v) {
  // ds_bpermute: gather across wave32 lanes; byte index
  return __int_as_float(__builtin_amdgcn_ds_bpermute(srcLane << 2, __float_as_int(v)));
}

// Raw hardware transcendentals (avoid IEEE guard/div-fixup expansions)
__device__ __forceinline__ float fast_rcp(float x) { return __builtin_amdgcn_rcpf(x); }
__device__ __forceinline__ float fast_rsq(float x) { return __builtin_amdgcn_rsqf(x); }

// ---------------------------------------------------------------------------
// Kernel 1: collision gradient via V_WMMA_F32_16X16X4_F32.
// One wave handles a 16-control-point tile x all 256 obstacles for one batch.
// Pairwise dot products are expanded into four K=4 (K=3 + zero pad) matmuls:
//   D1 = ref . obs          D2 = cp  . obs
//   D3 = ref^2 . (1/os^2)   D4 = ref . (obs/os^2)
// Then per-pair VALU:
//   n2 = ||ref-obs||^2, q = (cp-obs).(ref-obs), S = sum diff^2/os^2
//   dist = inv_n*(q - n2*rsqrt(S)),  de = CLEARANCE - dist
//   coef = -3*(min(dc,.5)^2 + (dc-min(dc,.5))), dc = max(de,0)   [branchless]
// cg[n] = ref[n]*sum_o(w) - sum_o(w*obs[o]),  w = coef/||diff||.
// ---------------------------------------------------------------------------
__global__ __launch_bounds__(32) void collision_grad_wmma(
    const float* __restrict__ cp,       // (B, NCP*3)
    const float* __restrict__ obs_loc,  // (B, NOBS, 3)
    const float* __restrict__ obs_size, // (B, NOBS, 3)
    const float* __restrict__ ref,      // (B, NCP, 3)
    float* __restrict__ cg)             // (B, NCP, 3) raw sums
{
  const int lane  = threadIdx.x;   // 0..31
  const int c     = lane & 15;     // column within tile / A-row within half
  const int h     = lane >> 4;     // half select (K01 vs K23)
  const int b     = blockIdx.y;
  const int ntile = blockIdx.x;
  const int row   = ntile * 16 + c;

  // ---- A-side loads (16 control points of this tile) ----
  const float* rp  = ref + ((size_t)b * NCP + row) * 3;
  const float rx = rp[0], ry = rp[1], rz = rp[2];
  const float* cpp = cp + ((size_t)b * NCP + row) * 3;
  const float cx = cpp[0], cy = cpp[1], cz = cpp[2];

  // A 16x4 f32 layout: v0 = K0 (lanes 0-15) / K2 (lanes 16-31); v1 = K1 / K3(pad)
  v2f refA, cpA, ref2A;
  if (h == 0) {
    refA  = (v2f){rx, ry};
    cpA   = (v2f){cx, cy};
    ref2A = (v2f){rx * rx, ry * ry};
  } else {
    refA  = (v2f){rz, 0.f};
    cpA   = (v2f){cz, 0.f};
    ref2A = (v2f){rz * rz, 0.f};
  }

  // Per-row scalars ||ref||^2 and cp.ref: combine the two K-halves (lane^16)
  float p_rn2 = refA.x * refA.x + refA.y * refA.y;
  float p_cpr = cpA.x * refA.x + cpA.y * refA.y;
  p_rn2 += bperm_f(lane ^ 16, p_rn2);
  p_cpr += bperm_f(lane ^ 16, p_cpr);

  // Lane L now holds the scalars of row (L&15). C/D layout needs rows v+8h.
  float rn2r[8], cprr[8];
#pragma unroll
  for (int v = 0; v < 8; ++v) {
    rn2r[v] = bperm_f(v + 8 * h, p_rn2);
    cprr[v] = bperm_f(v + 8 * h, p_cpr);
  }

  float aw[8], awx[8], awy[8], awz[8];
#pragma unroll
  for (int v = 0; v < 8; ++v) { aw[v] = awx[v] = awy[v] = awz[v] = 0.f; }

  for (int ot = 0; ot < NOBS / 16; ++ot) {
    const int o = ot * 16 + c;
    const float* op = obs_loc  + ((size_t)b * NOBS + o) * 3;
    const float ox = op[0], oy = op[1], oz = op[2];
    const float* sp = obs_size + ((size_t)b * NOBS + o) * 3;
    const float sx = sp[0], sy = sp[1], sz = sp[2];
    const float ix = fast_rcp(sx * sx);
    const float iy = fast_rcp(sy * sy);
    const float iz = fast_rcp(sz * sz);

    // B 4x16 f32 layout mirrors A: v0 = K0/K2, v1 = K1/K3(pad)
    v2f obsB, ios2B, oiB;
    if (h == 0) {
      obsB  = (v2f){ox, oy};
      ios2B = (v2f){ix, iy};
      oiB   = (v2f){ox * ix, oy * iy};
    } else {
      obsB  = (v2f){oz, 0.f};
      ios2B = (v2f){iz, 0.f};
      oiB   = (v2f){oz * iz, 0.f};
    }
    const float on2 = ox * ox + oy * oy + oz * oz;
    const float oS  = ox * ox * ix + oy * oy * iy + oz * oz * iz;

    v8f zero = {};
    v8f d1 = __builtin_amdgcn_wmma_f32_16x16x4_f32(false, refA,  false, obsB,
                                                   (short)0, zero, false, false);
    v8f d2 = __builtin_amdgcn_wmma_f32_16x16x4_f32(false, cpA,   false, obsB,
                                                   (short)0, zero, false, false);
    v8f d3 = __builtin_amdgcn_wmma_f32_16x16x4_f32(false, ref2A, false, ios2B,
                                                   (short)0, zero, false, false);
    v8f d4 = __builtin_amdgcn_wmma_f32_16x16x4_f32(false, refA,  false, oiB,
                                                   (short)0, zero, false, false);

    // C/D layout: lane holds rows v+8h (v=0..7), column == c (same obs we loaded)
#pragma unroll
    for (int v = 0; v < 8; ++v) {
      const float m1 = d1[v], m2 = d2[v], m3 = d3[v], m4 = d4[v];
      const float n2 = rn2r[v] - 2.f * m1 + on2;          // ||ref-obs||^2
      const float q  = cprr[v] - m2 - m1 + on2;           // (cp-obs).(ref-obs)
      const float S  = m3 - 2.f * m4 + oS;                // sum diff^2/os^2
      const float inv_n = fast_rsq(n2);                   // v_rsq_f32
      // dist = q*inv_n - sqrt(n2)*rsqrt(S) == inv_n*(q - n2*rsqrt(S))
      const float dist = inv_n * (q - n2 * fast_rsq(S));
      const float de = 0.5f - dist;                       // CLEARANCE - dist
      // branchless piecewise (linear part is the tangent continuation at 0.5):
      const float dc = fmaxf(de, 0.f);
      const float dq = fminf(dc, 0.5f);
      const float coef = -3.f * (dq * dq + (dc - dq));
      const float w = coef * inv_n;
      aw[v]  += w;
      awx[v] += w * ox;
      awy[v] += w * oy;
      awz[v] += w * oz;
    }
  }

  // Reduce each accumulator across the 16 lanes of each half-wave
#pragma unroll
  for (int m = 1; m < 16; m <<= 1) {
#pragma unroll
    for (int v = 0; v < 8; ++v) {
      aw[v]  += bperm_f(lane ^ m, aw[v]);
      awx[v] += bperm_f(lane ^ m, awx[v]);
      awy[v] += bperm_f(lane ^ m, awy[v]);
      awz[v] += bperm_f(lane ^ m, awz[v]);
    }
  }

  // One writer lane per row: lane with c==v in half h writes row v+8h
#pragma unroll
  for (int v = 0; v < 8; ++v) {
    if (c == v) {
      const int rowg = ntile * 16 + v + 8 * h;
      const float* rpo = ref + ((size_t)b * NCP + rowg) * 3;
      float* po = cg + ((size_t)b * NCP + rowg) * 3;
      po[0] = rpo[0] * aw[v] - awx[v];
      po[1] = rpo[1] * aw[v] - awy[v];
      po[2] = rpo[2] * aw[v] - awz[v];
    }
  }
}

// ---------------------------------------------------------------------------
// Kernel 2: jerk smoothness + velocity/acceleration feasibility stencils,
// combined with the collision gradient from d_ws. One thread per (b, n).
// ---------------------------------------------------------------------------
__global__ __launch_bounds__(256) void smooth_feas_combine(
    const float* __restrict__ cp,   // (B, NCP*3)
    const float* __restrict__ cg,   // (B, NCP, 3) raw sums
    float* __restrict__ out)        // (B, NCP*3)
{
  const int idx = blockIdx.x * blockDim.x + threadIdx.x;
  if (idx >= BATCH * NCP) return;
  const int b = idx / NCP;
  const int n = idx % NCP;

  // window of control points n-3 .. n+3 (zeros outside; masked anyway)
  float w[7][3];
#pragma unroll
  for (int t = 0; t < 7; ++t) {
    const int j = n - 3 + t;
    if (j >= 0 && j < NCP) {
      const float* p = cp + ((size_t)b * NCP + j) * 3;
      w[t][0] = p[0]; w[t][1] = p[1]; w[t][2] = p[2];
    } else {
      w[t][0] = w[t][1] = w[t][2] = 0.f;
    }
  }

  const float invB = 1.f / (float)BATCH;
  float res[3];
#pragma unroll
  for (int k = 0; k < 3; ++k) {
    // smoothness: sg[n] = -2 J(n) + 6 J(n-1) - 6 J(n-2) + 2 J(n-3), J valid [0,NCP-4]
    float sg = 0.f;
#pragma unroll
    for (int t = -3; t <= 0; ++t) {
      const int j = n + t;
      if (j >= 0 && j <= NCP - 4) {
        const int i = t + 3;
        const float J = w[i + 3][k] - 3.f * w[i + 2][k] + 3.f * w[i + 1][k] - w[i][k];
        const float wt = (t == 0) ? -2.f : (t == -1) ? 6.f : (t == -2) ? -6.f : 2.f;
        sg += wt * J;
      }
    }
    sg *= invB;

    // velocity feasibility: gv(i) = 2*(|vel|/dt - 3)/dt if over limit; dt = 0.1
    float vfg = 0.f;
    if (n <= NCP - 2) {
      const float velk = fabsf(w[4][k] - w[3][k]) * 10.f;
      vfg -= (velk >= 3.f) ? 2.f * (velk - 3.f) * 10.f : 0.f;
    }
    if (n >= 1) {
      const float velk = fabsf(w[3][k] - w[2][k]) * 10.f;
      vfg += (velk >= 3.f) ? 2.f * (velk - 3.f) * 10.f : 0.f;
    }

    // acceleration feasibility: ga(i) = 2*(|acc|/dt^2 - 6)/dt^2 if over limit
    float afg = 0.f;
    if (n <= NCP - 3) {
      const float ak = fabsf(w[5][k] - 2.f * w[4][k] + w[3][k]) * 100.f;
      afg += (ak >= 6.f) ? 2.f * (ak - 6.f) * 100.f : 0.f;
    }
    if (n >= 1 && n <= NCP - 2) {
      const float ak = fabsf(w[4][k] - 2.f * w[3][k] + w[2][k]) * 100.f;
      afg -= (ak >= 6.f) ? 2.f * (2.f * (ak - 6.f) * 100.f) : 0.f;
    }
    if (n >= 2) {
      const float ak = fabsf(w[3][k] - 2.f * w[2][k] + w[1][k]) * 100.f;
      afg += (ak >= 6.f) ? 2.f * (ak - 6.f) * 100.f : 0.f;
    }
    const float fg = vfg * 100.f + afg;   // vfg/dt^2 + afg

    const float cgv = cg[(size_t)idx * 3 + k] * invB;
    res[k] = -0.01f * (1.f * sg + 10.f * cgv + 1.f * fg);  // -LR*(LAM_S*sg+LAM_C*cg+LAM_F*fg)
  }
  float* po = out + (size_t)idx * 3;
  po[0] = res[0]; po[1] = res[1]; po[2] = res[2];
}

extern "C" void kernel_launch(void* const* d_in, const int* in_sizes, int n_in,
                              void* d_out, int out_size, void* d_ws, size_t ws_size,
                              hipStream_t stream) {
  (void)in_sizes; (void)n_in; (void)out_size; (void)ws_size;
  // inputs: [0]=t (unused), [1]=control_pts, [2]=obs_loc, [3]=obs_size, [4]=reference_pts
  const float* cp      = (const float*)d_in[1];
  const float* obs_loc = (const float*)d_in[2];
  const float* obs_siz = (const float*)d_in[3];
  const float* ref     = (const float*)d_in[4];
  float* out = (float*)d_out;
  float* cg  = (float*)d_ws;   // B*NCP*3 floats = 192 KB scratch

  dim3 g1(NCP / 16, BATCH);    // 32 x 32 waves
  collision_grad_wmma<<<g1, 32, 0, stream>>>(cp, obs_loc, obs_siz, ref, cg);

  const int total = BATCH * NCP;
  smooth_feas_combine<<<(total + 255) / 256, 256, 0, stream>>>(cp, cg, out);
}